// Attention_69861938037646
// MI455X (gfx1250) — compile-verified
//
#include <hip/hip_runtime.h>
#include <math.h>
#include <stdint.h>

typedef float v2f __attribute__((ext_vector_type(2)));
typedef float v8f __attribute__((ext_vector_type(8)));

#define NS 200
#define NE 32
#define NTILE 13          // ceil(200/16)
#define NSP (NTILE * 16)  // 208 padded rows

__device__ __forceinline__ float fast_sigmoid(float x) {
    // v_exp_f32 + v_rcp_f32, avoids the IEEE divide expansion
    return __builtin_amdgcn_rcpf(1.f + __expf(-x));
}

__global__ __launch_bounds__(256)
void attn_fused_kernel(const float* __restrict__ all_memory,
                       const float* __restrict__ last_memory,
                       const float* __restrict__ average_memory,
                       const float* __restrict__ W1,
                       const float* __restrict__ b1,
                       const float* __restrict__ W2,
                       const float* __restrict__ b2,
                       const float* __restrict__ W3,
                       const float* __restrict__ b3,
                       const float* __restrict__ W4,
                       float* __restrict__ out)
{
    __shared__ float As[NSP * NE];      // padded A tile (rows 200..207 zeroed)
    __shared__ float vbias[NE];         // per-batch bias vector
    __shared__ float scores[NSP];       // sigmoid(proj) @ W4 per row
    __shared__ float outred[8][NE];     // final 8-way reduction

    const int b    = blockIdx.x;
    const int tid  = threadIdx.x;
    const int lane = tid & 31;
    const int wave = tid >> 5;
    const int n    = lane & 15;   // WMMA lane -> row (A) / column (B,C)
    const int half = lane >> 4;   // WMMA K/M half selector

    // ---- Phase 1: stage all_memory[b] into LDS via async global->LDS DMA ----
    {
        const float* gsrc = all_memory + (size_t)b * NS * NE;
        for (int i = tid; i < (NS * NE) / 4; i += 256) {
            uint32_t ldsaddr = (uint32_t)(uintptr_t)&As[i * 4];  // LDS byte offset
            uint32_t goff    = (uint32_t)(i * 16);               // global byte offset
            asm volatile("global_load_async_to_lds_b128 %0, %1, %2"
                         :: "v"(ldsaddr), "v"(goff), "s"(gsrc)
                         : "memory");
        }
        // zero-pad rows 200..207 (plain DS stores, covered by the barrier)
        const float4 z4 = make_float4(0.f, 0.f, 0.f, 0.f);
        float4* dst = (float4*)As;
        for (int i = (NS * NE) / 4 + tid; i < (NSP * NE) / 4; i += 256) dst[i] = z4;
        asm volatile("s_wait_asynccnt 0" ::: "memory");
    }

    // ---- Phase 2: bias vector v = b1 + last@W2 + b2 + avg@W3 + b3 ----
    if (tid < NE) {
        float acc = b1[tid] + b2[tid] + b3[tid];
        const float* lm = last_memory    + (size_t)b * NE;
        const float* am = average_memory + (size_t)b * NE;
        #pragma unroll
        for (int k = 0; k < NE; ++k)
            acc += lm[k] * W2[k * NE + tid] + am[k] * W3[k * NE + tid];
        vbias[tid] = acc;
    }
    __syncthreads();

    // ---- Phase 3: W1 B-fragments in registers (B 4x16 layout: lane=N, K split by half) ----
    v2f Bf0[8], Bf1[8];
    #pragma unroll
    for (int k = 0; k < 8; ++k) {
        const int kr = 4 * k + 2 * half;
        Bf0[k].x = W1[kr * NE + n];           Bf0[k].y = W1[(kr + 1) * NE + n];
        Bf1[k].x = W1[kr * NE + n + 16];      Bf1[k].y = W1[(kr + 1) * NE + n + 16];
    }
    const float w4a = W4[n];
    const float w4b = W4[n + 16];
    const float vb0 = vbias[n];
    const float vb1 = vbias[n + 16];

    // ---- Phase 4: per-wave 16-row tiles: proj = A@W1 (fp32 WMMA), sigmoid, dot W4 ----
    for (int t = wave; t < NTILE; t += 8) {
        const int s0 = t * 16;
        v8f acc0 = {0.f, 0.f, 0.f, 0.f, 0.f, 0.f, 0.f, 0.f};
        v8f acc1 = {0.f, 0.f, 0.f, 0.f, 0.f, 0.f, 0.f, 0.f};
        #pragma unroll
        for (int k = 0; k < 8; ++k) {
            // A 16x4 f32 layout: lane -> row = lane%16; half picks K pair within slab
            const int row = s0 + n;
            const int col = 4 * k + 2 * half;
            v2f a = *(const v2f*)&As[row * NE + col];
            acc0 = __builtin_amdgcn_wmma_f32_16x16x4_f32(
                       false, a, false, Bf0[k], (short)0, acc0, false, false);
            acc1 = __builtin_amdgcn_wmma_f32_16x16x4_f32(
                       false, a, false, Bf1[k], (short)0, acc1, false, false);
        }
        // C layout: lane gives N (= n, n+16), VGPR r gives M = r + 8*half
        float tsum[8];
        #pragma unroll
        for (int r = 0; r < 8; ++r) {
            const float g0 = fast_sigmoid(acc0[r] + vb0);
            const float g1 = fast_sigmoid(acc1[r] + vb1);
            tsum[r] = g0 * w4a + g1 * w4b;   // this lane's 2 columns of the W4 dot
        }
        // reduce over the 16 lanes of each half (covers all 32 columns)
        #pragma unroll
        for (int mask = 1; mask <= 8; mask <<= 1) {
            #pragma unroll
            for (int r = 0; r < 8; ++r)
                tsum[r] += __shfl_xor(tsum[r], mask, 32);
        }
        if (n == 0) {
            #pragma unroll
            for (int r = 0; r < 8; ++r)
                scores[s0 + 8 * half + r] = tsum[r];
        }
    }
    __syncthreads();

    // ---- Phase 5: out[b][e] = sum_s scores[s] * A[s][e] (padded rows are zero) ----
    {
        const int e  = tid & 31;
        const int sg = tid >> 5;
        float acc = 0.f;
        for (int s = sg; s < NSP; s += 8)
            acc += scores[s] * As[s * NE + e];
        outred[sg][e] = acc;
    }
    __syncthreads();
    if (tid < NE) {
        float r = 0.f;
        #pragma unroll
        for (int w = 0; w < 8; ++w) r += outred[w][tid];
        out[(size_t)b * NE + tid] = r;
    }
}

extern "C" void kernel_launch(void* const* d_in, const int* in_sizes, int n_in,
                              void* d_out, int out_size, void* d_ws, size_t ws_size,
                              hipStream_t stream) {
    // setup_inputs order:
    // 0 all_memory [B,S,E] f32   1 last_memory [B,E] f32   2 average_memory [B,E] f32
    // 3 mask [B,S] bool (unused) 4 W1 [E,E]  5 b1 [E]  6 W2 [E,E]  7 b2 [E]
    // 8 W3 [E,E]  9 b3 [E]  10 W4 [E,1]
    const float* all_memory     = (const float*)d_in[0];
    const float* last_memory    = (const float*)d_in[1];
    const float* average_memory = (const float*)d_in[2];
    const float* W1 = (const float*)d_in[4];
    const float* b1 = (const float*)d_in[5];
    const float* W2 = (const float*)d_in[6];
    const float* b2 = (const float*)d_in[7];
    const float* W3 = (const float*)d_in[8];
    const float* b3 = (const float*)d_in[9];
    const float* W4 = (const float*)d_in[10];
    float* out = (float*)d_out;

    const int nbatch = in_sizes[0] / (NS * NE);   // = 4096
    attn_fused_kernel<<<nbatch, 256, 0, stream>>>(
        all_memory, last_memory, average_memory,
        W1, b1, W2, b2, W3, b3, W4, out);
}